// CartographerPoseCorrector_84928683311338
// MI455X (gfx1250) — compile-verified
//
#include <hip/hip_runtime.h>
#include <stdint.h>

typedef __attribute__((ext_vector_type(2))) float v2f;
typedef __attribute__((ext_vector_type(8))) float v8f;

#define HW        128
#define NPIX      (HW * HW)
#define MAXP      8
#define PAIR_WS   32768                    // bytes of workspace per pair
#define META_SZ   256
#define NC_COARSE (21 * 21 * 16)           // 7056
#define NC_FINE   (9 * 9 * 9)              // 729
#define OFF_FINE  (NC_COARSE * 4)          // 28224
#define OFF_CP    (OFF_FINE + NC_FINE * 4) // 31140
#define THRESH_V  0.3f
#define DEG2RAD   0.017453292519943295f

struct PairMeta { int ego, nbr, b, n; };

__device__ __forceinline__ float thr(float v) { return v > THRESH_V ? v : 0.0f; }

__device__ __forceinline__ void make_aff(const float* bA, float dx, float dy,
                                         float dr, float* o) {
  float th_ = dr * DEG2RAD;
  float cn = cosf(th_), sn = sinf(th_);
  float tx = dx * (2.0f / (HW - 1));
  float ty = dy * (2.0f / (HW - 1));
  o[0] =  bA[0] * cn + bA[1] * sn;
  o[1] = -bA[0] * sn + bA[1] * cn;
  o[2] =  bA[0] * tx + bA[1] * ty + bA[2];
  o[3] =  bA[3] * cn + bA[4] * sn;
  o[4] = -bA[3] * sn + bA[4] * cn;
  o[5] =  bA[3] * tx + bA[4] * ty + bA[5];
}

// ---------------------------------------------------------------------------
// Init: build pair list from record_len, copy base affines into the output.
// ---------------------------------------------------------------------------
__global__ __launch_bounds__(256)
void pc_init_kernel(const int* __restrict__ rl_raw, int B, int Mtot, int maxPairs,
                    const float* __restrict__ affine, int out_n,
                    float* __restrict__ out, char* __restrict__ ws) {
  for (int i = threadIdx.x; i < out_n; i += 256) out[i] = affine[i];
  if (threadIdx.x == 0) {
    // record_len is np.int64 in the reference; detect int64 (odd word zero)
    // vs a harness-downcast int32 layout.
    bool is64 = (B >= 1) && (rl_raw[0] > 0) && (rl_raw[1] == 0);
    PairMeta* pm = (PairMeta*)(ws + 16);
    int idx = 0, np = 0;
    for (int b = 0; b < B; ++b) {
      int N = is64 ? rl_raw[2 * b] : rl_raw[b];
      if (N < 0) N = 0;
      for (int n = 1; n < N; ++n) {
        if (np < maxPairs && idx < Mtot && (idx + n) < Mtot) {
          pm[np].ego = idx; pm[np].nbr = idx + n; pm[np].b = b; pm[np].n = n;
          ++np;
        }
      }
      idx += N;
    }
    ((int*)ws)[0] = np;
  }
}

// ---------------------------------------------------------------------------
// Candidate scoring. mode 0: coarse 21x21x16 grid; mode 1: fine 9x9x9 around
// the coarse winner. Each wave scores 16 candidates over all 16384 pixels by
// chaining v_wmma_f32_16x16x4_f32: A = warped nbr samples (16 cands x 4 px),
// B = ego pixel values broadcast across N, C accumulates scores.
// Both images live in LDS (async-staged, pre-thresholded): the inner loop is
// pure DS + VALU + WMMA with no VMEM stalls.
// ---------------------------------------------------------------------------
__global__ __launch_bounds__(256)
void pc_score_kernel(const float* __restrict__ occ,
                     const float* __restrict__ affine,
                     char* __restrict__ ws, int mode,
                     const unsigned char* __restrict__ acp) {
  __shared__ float s_nbr[NPIX];
  __shared__ float s_ego[NPIX];

  const int pid = blockIdx.y;
  const int np  = ((const int*)ws)[0];
  if (pid >= np) return;                       // block-uniform exit

  PairMeta pmv = ((const PairMeta*)(ws + 16))[pid];
  const int egoIdx = __builtin_amdgcn_readfirstlane(pmv.ego);
  const int nbrIdx = __builtin_amdgcn_readfirstlane(pmv.nbr);
  const int bIdx   = __builtin_amdgcn_readfirstlane(pmv.b);
  const int nIdx   = __builtin_amdgcn_readfirstlane(pmv.n);

  char* pws = ws + META_SZ + (size_t)pid * PAIR_WS;
  float* scores   = (float*)(pws + (mode ? OFF_FINE : 0));
  const float* cp = (const float*)(pws + OFF_CP);
  const int Ncand = mode ? NC_FINE : NC_COARSE;

  const float* nbr_g = occ + (size_t)nbrIdx * NPIX;
  const float* ego_g = occ + (size_t)egoIdx * NPIX;

  // ---- CDNA5 async global->LDS staging of both maps ----
  {
    unsigned nbr_base = (unsigned)(uintptr_t)(&s_nbr[0]);
    unsigned ego_base = (unsigned)(uintptr_t)(&s_ego[0]);
    const int t = threadIdx.x;
#pragma unroll
    for (int i = 0; i < 16; ++i) {
      int chunk = t + i * 256;                 // 4096 x 16B chunks per image
      unsigned off = (unsigned)chunk * 16u;
      unsigned ln = nbr_base + off;
      unsigned le = ego_base + off;
      asm volatile("global_load_async_to_lds_b128 %0, %1, %2"
                   :: "v"(ln), "v"(off), "s"(nbr_g) : "memory");
      asm volatile("global_load_async_to_lds_b128 %0, %1, %2"
                   :: "v"(le), "v"(off), "s"(ego_g) : "memory");
    }
    asm volatile("s_wait_asynccnt 0x0" ::: "memory");
    __syncthreads();
    for (int i = 0; i < 64; ++i) {             // threshold both in place
      int idx = t + i * 256;
      s_nbr[idx] = thr(s_nbr[idx]);
      s_ego[idx] = thr(s_ego[idx]);
    }
    __syncthreads();
  }

  const int wave = threadIdx.x >> 5;
  const int lane = threadIdx.x & 31;
  const int candBase = (blockIdx.x * 8 + wave) * 16;
  if (candBase >= Ncand) return;               // wave-uniform exit

  int ci = candBase + (lane & 15);
  if (ci >= Ncand) ci = Ncand - 1;             // clamp: keep EXEC all-ones

  float dx, dy, dr;
  if (mode == 0) {
    int idr = ci & 15, t2 = ci >> 4;
    int idy = t2 % 21, idx_ = t2 / 21;
    dx = -20.f + 2.f * (float)idx_;
    dy = -20.f + 2.f * (float)idy;
    dr = -15.f + 2.f * (float)idr;
  } else {
    int idr = ci % 9, t2 = ci / 9;
    int idy = t2 % 9, idx_ = t2 / 9;
    dx = cp[0] - 2.f + 0.5f * (float)idx_;
    dy = cp[1] - 2.f + 0.5f * (float)idy;
    dr = cp[2] - 2.f + 0.5f * (float)idr;
  }

  // grid / sampling constants (align_corners flag read from device scalar)
  int ac = (acp != nullptr) && (acp[0] != 0);
  float gsc, gof, isc, iof;
  if (ac) { gsc = 2.f / (HW - 1); gof = -1.f;
            isc = 0.5f * (HW - 1); iof = 0.5f * (HW - 1); }
  else    { gsc = 2.f / HW; gof = 1.f / HW - 1.f;
            isc = 0.5f * HW; iof = 0.5f * HW - 0.5f; }

  const float* bA = affine + (bIdx * 4 + nIdx) * 6;
  float aff[6];
  make_aff(bA, dx, dy, dr, aff);
  // fold grid->pixel mapping into the affine
  float P00 = isc * aff[0], P01 = isc * aff[1], P02 = isc * aff[2] + iof;
  float P10 = isc * aff[3], P11 = isc * aff[4], P12 = isc * aff[5] + iof;

  auto tap = [&](float xf, float yf) -> float {
    bool valid = (xf >= 0.f) && (xf <= (float)(HW - 1)) &&
                 (yf >= 0.f) && (yf <= (float)(HW - 1));
    float xc = fminf(fmaxf(xf, 0.f), (float)(HW - 1));
    float yc = fminf(fmaxf(yf, 0.f), (float)(HW - 1));
    int xi = (int)xc, yi = (int)yc;
    float v = s_nbr[(yi << 7) + xi];
    return valid ? v : 0.f;
  };
  auto sample = [&](int p) -> float {
    int w = p & (HW - 1), h = p >> 7;
    float gx = (float)w * gsc + gof;
    float gy = (float)h * gsc + gof;
    float ix = P00 * gx + P01 * gy + P02;
    float iy = P10 * gx + P11 * gy + P12;
    float x0 = floorf(ix), y0 = floorf(iy);
    float wx = ix - x0, wy = iy - y0;
    float t00 = tap(x0,       y0);
    float t10 = tap(x0 + 1.f, y0);
    float t01 = tap(x0,       y0 + 1.f);
    float t11 = tap(x0 + 1.f, y0 + 1.f);
    float top = (1.f - wx) * t00 + wx * t10;
    float bot = (1.f - wx) * t01 + wx * t11;
    return (1.f - wy) * top + wy * bot;
  };

  // A-matrix 16x4 f32 layout: lanes 0-15 carry K=0,1 ; lanes 16-31 carry K=2,3
  const int kHalf = (lane >> 4) << 1;
  v8f acc = {0.f, 0.f, 0.f, 0.f, 0.f, 0.f, 0.f, 0.f};
  for (int kb = 0; kb < NPIX; kb += 8) {       // 2 WMMAs per trip (K=8 pixels)
    int p0 = kb + kHalf;
    v2f bm0 = *(const v2f*)(&s_ego[p0]);       // broadcast ds_load_b64
    v2f am0; am0.x = sample(p0); am0.y = sample(p0 + 1);
    int p1 = kb + 4 + kHalf;
    v2f bm1 = *(const v2f*)(&s_ego[p1]);
    v2f am1; am1.x = sample(p1); am1.y = sample(p1 + 1);
    acc = __builtin_amdgcn_wmma_f32_16x16x4_f32(false, am0, false, bm0,
                                                (short)0, acc, false, false);
    acc = __builtin_amdgcn_wmma_f32_16x16x4_f32(false, am1, false, bm1,
                                                (short)0, acc, false, false);
  }

  // D layout: lane 0 col holds M=0..7 in VGPR 0..7; lane 16 holds M=8..15.
  if (lane == 0) {
#pragma unroll
    for (int j = 0; j < 8; ++j) {
      int c2 = candBase + j;
      if (c2 < Ncand) scores[c2] = acc[j];
    }
  } else if (lane == 16) {
#pragma unroll
    for (int j = 0; j < 8; ++j) {
      int c2 = candBase + 8 + j;
      if (c2 < Ncand) scores[c2] = acc[j];
    }
  }
}

// ---------------------------------------------------------------------------
// Coarse argmax (first-max semantics like jnp.argmax) -> coarse params cp.
// ---------------------------------------------------------------------------
__global__ __launch_bounds__(256)
void pc_reduce_kernel(char* __restrict__ ws) {
  __shared__ float s_v[256];
  __shared__ int   s_i[256];
  const int pid = blockIdx.x;
  if (pid >= ((const int*)ws)[0]) return;
  char* pws = ws + META_SZ + (size_t)pid * PAIR_WS;
  const float* scores = (const float*)pws;

  float bv = -__builtin_inff(); int bi = 0;
  for (int c = threadIdx.x; c < NC_COARSE; c += 256) {
    float v = scores[c];
    if (v > bv) { bv = v; bi = c; }
  }
  s_v[threadIdx.x] = bv; s_i[threadIdx.x] = bi;
  __syncthreads();
  for (int off = 128; off > 0; off >>= 1) {
    if (threadIdx.x < off) {
      float ov = s_v[threadIdx.x + off]; int oi = s_i[threadIdx.x + off];
      if (ov > s_v[threadIdx.x] ||
          (ov == s_v[threadIdx.x] && oi < s_i[threadIdx.x])) {
        s_v[threadIdx.x] = ov; s_i[threadIdx.x] = oi;
      }
    }
    __syncthreads();
  }
  if (threadIdx.x == 0) {
    float best = s_v[0]; int idx = s_i[0];
    bool ok = best > 1e-5f;
    int idr = idx & 15, t2 = idx >> 4;
    int idy = t2 % 21, idx_ = t2 / 21;
    float* cp = (float*)(pws + OFF_CP);
    cp[0] = ok ? (-20.f + 2.f * (float)idx_) : 0.f;
    cp[1] = ok ? (-20.f + 2.f * (float)idy)  : 0.f;
    cp[2] = ok ? (-15.f + 2.f * (float)idr)  : 0.f;
  }
}

// ---------------------------------------------------------------------------
// Fine argmax + final affine with the reference's fallback logic.
// ---------------------------------------------------------------------------
__global__ __launch_bounds__(256)
void pc_final_kernel(const float* __restrict__ affine, char* __restrict__ ws,
                     float* __restrict__ out) {
  __shared__ float s_v[256];
  __shared__ int   s_i[256];
  const int pid = blockIdx.x;
  if (pid >= ((const int*)ws)[0]) return;
  PairMeta m = ((const PairMeta*)(ws + 16))[pid];
  char* pws = ws + META_SZ + (size_t)pid * PAIR_WS;
  const float* scores = (const float*)(pws + OFF_FINE);
  const float* cp     = (const float*)(pws + OFF_CP);

  float bv = -__builtin_inff(); int bi = 0;
  for (int c = threadIdx.x; c < NC_FINE; c += 256) {
    float v = scores[c];
    if (v > bv) { bv = v; bi = c; }
  }
  s_v[threadIdx.x] = bv; s_i[threadIdx.x] = bi;
  __syncthreads();
  for (int off = 128; off > 0; off >>= 1) {
    if (threadIdx.x < off) {
      float ov = s_v[threadIdx.x + off]; int oi = s_i[threadIdx.x + off];
      if (ov > s_v[threadIdx.x] ||
          (ov == s_v[threadIdx.x] && oi < s_i[threadIdx.x])) {
        s_v[threadIdx.x] = ov; s_i[threadIdx.x] = oi;
      }
    }
    __syncthreads();
  }
  if (threadIdx.x == 0) {
    int aoff = (m.b * 4 + m.n) * 6;
    const float* bA = affine + aoff;
    float best = s_v[0]; int idx = s_i[0];
    bool cpzero = (cp[0] == 0.f) && (cp[1] == 0.f) && (cp[2] == 0.f);
    bool ok = best > 1e-5f;
    float o[6];
    if (!ok || cpzero) {
      for (int j = 0; j < 6; ++j) o[j] = bA[j];
    } else {
      int idr = idx % 9, t2 = idx / 9;
      int idy = t2 % 9, idx_ = t2 / 9;
      float dx = cp[0] - 2.f + 0.5f * (float)idx_;
      float dy = cp[1] - 2.f + 0.5f * (float)idy;
      float dr = cp[2] - 2.f + 0.5f * (float)idr;
      make_aff(bA, dx, dy, dr, o);
    }
    for (int j = 0; j < 6; ++j) out[aoff + j] = o[j];
  }
}

// ---------------------------------------------------------------------------
extern "C" void kernel_launch(void* const* d_in, const int* in_sizes, int n_in,
                              void* d_out, int out_size, void* d_ws, size_t ws_size,
                              hipStream_t stream) {
  const float* occ    = (const float*)d_in[0];
  const int*   rl     = (const int*)d_in[1];
  const float* affine = (const float*)d_in[2];
  const unsigned char* acflag =
      (n_in > 3) ? (const unsigned char*)d_in[3] : nullptr;
  float* out = (float*)d_out;
  char*  ws  = (char*)d_ws;

  int Mtot = in_sizes[0] / NPIX;                 // number of 128x128 maps
  int B    = (n_in > 1) ? in_sizes[1] : 0;       // number of batches
  int maxPairs = (Mtot < MAXP) ? Mtot : MAXP;
  if (ws_size < (size_t)META_SZ) return;
  size_t cap = (ws_size - META_SZ) / PAIR_WS;
  if ((size_t)maxPairs > cap) maxPairs = (int)cap;

  pc_init_kernel<<<1, 256, 0, stream>>>(rl, B, Mtot, maxPairs, affine,
                                        out_size, out, ws);
  if (maxPairs < 1) return;

  dim3 gC(((NC_COARSE / 16) + 7) / 8, maxPairs);      // 441 waves -> 56 blocks
  pc_score_kernel<<<gC, 256, 0, stream>>>(occ, affine, ws, 0, acflag);
  pc_reduce_kernel<<<maxPairs, 256, 0, stream>>>(ws);
  dim3 gF((((NC_FINE + 15) / 16) + 7) / 8, maxPairs); // 46 waves -> 6 blocks
  pc_score_kernel<<<gF, 256, 0, stream>>>(occ, affine, ws, 1, acflag);
  pc_final_kernel<<<maxPairs, 256, 0, stream>>>(affine, ws, out);
}